// ClusterNetHetero_74947179315777
// MI455X (gfx1250) — compile-verified
//
#include <hip/hip_runtime.h>

#define NODES   100000
#define EDGES   3200000
#define DIM     128
#define NGRAPH  64
#define NCLS    4
#define NLAYER  3
#define LDS_STRIDE 130   /* 128 + 2 pad DWORDs: bank-conflict-free, 8B aligned */

typedef float v2f __attribute__((ext_vector_type(2)));
typedef float v8f __attribute__((ext_vector_type(8)));
typedef unsigned int v4u __attribute__((ext_vector_type(4)));
typedef int v4i __attribute__((ext_vector_type(4)));
typedef int v8i __attribute__((ext_vector_type(8)));

// Fold Linear bias + BatchNorm(eval): out = (hW)*alpha + shift
__global__ __launch_bounds__(256) void k_affine(const float* __restrict__ bs,
                                                const float* __restrict__ gam,
                                                const float* __restrict__ bet,
                                                const float* __restrict__ rm,
                                                const float* __restrict__ rv,
                                                float* __restrict__ alpha,
                                                float* __restrict__ shift) {
  int i = blockIdx.x * blockDim.x + threadIdx.x;
  if (i >= NLAYER * 2 * DIM) return;
  float s = gam[i] * rsqrtf(rv[i] + 1e-5f);
  alpha[i] = s;
  shift[i] = (bs[i] - rm[i]) * s + bet[i];
}

// Permute all 6 weight matrices into WMMA B-fragment order:
// wp[mat][k0i][lane][j][pair] = W[mat][(4*k0i + 2*(lane>>4) + pair)*128 + j*16 + (lane&15)]
// -> each lane's per-k-step B data is one contiguous 64B run (4x global_load_b128).
__global__ __launch_bounds__(256) void k_packW(const float* __restrict__ Ws,
                                               float* __restrict__ wp) {
  int idx = blockIdx.x * blockDim.x + threadIdx.x;
  if (idx >= NLAYER * 2 * 32 * 32 * 8 * 2) return;
  int mat  = idx >> 14;
  int r    = idx & 16383;
  int pair = r & 1;
  int j    = (r >> 1) & 7;
  int lane = (r >> 4) & 31;
  int k0i  = r >> 9;
  int ka   = k0i * 4 + (lane >> 4) * 2 + pair;
  int col  = j * 16 + (lane & 15);
  wp[idx] = Ws[(size_t)mat * DIM * DIM + ka * DIM + col];
}

__global__ __launch_bounds__(256) void k_copy4(const float4* __restrict__ in,
                                               float4* __restrict__ out, int n4) {
  int i = blockIdx.x * blockDim.x + threadIdx.x;
  if (i < n4) out[i] = in[i];
}

// One wave per edge; lanes cover the 128-float row. buf pre-init to h,
// so afterwards buf = h + segment_sum(h[src], dst).
__global__ __launch_bounds__(256) void k_scatter(const float* __restrict__ h,
                                                 float* __restrict__ buf,
                                                 const int* __restrict__ src,
                                                 const int* __restrict__ dst) {
  int e = blockIdx.x * 8 + (threadIdx.x >> 5);
  int lane = threadIdx.x & 31;
  if (e >= EDGES) return;
  int s = src[e], d = dst[e];
  const float4 v = *(const float4*)(h + (size_t)s * DIM + lane * 4);
  float* p = buf + (size_t)d * DIM + lane * 4;
  atomicAdd(p + 0, v.x);
  atomicAdd(p + 1, v.y);
  atomicAdd(p + 2, v.z);
  atomicAdd(p + 3, v.w);
}

__device__ __forceinline__ v8f wmma_f32(v2f a, v2f b, v8f c) {
  return __builtin_amdgcn_wmma_f32_16x16x4_f32(false, a, false, b, (short)0, c,
                                               false, false);
}

// Fused 2-layer MLP for one 16-row tile, in place. One wave32, EXEC all-ones.
// A tile arrives via the Tensor Data Mover (TDM) into padded LDS; both GEMMs
// read A-fragments from LDS; B-fragments come from the packed weight layout.
__global__ __launch_bounds__(32) void k_mlp(float* __restrict__ hbuf,
                                            const float* __restrict__ W1p,
                                            const float* __restrict__ W2p,
                                            const float* __restrict__ a1,
                                            const float* __restrict__ s1,
                                            const float* __restrict__ a2,
                                            const float* __restrict__ s2) {
  __shared__ float ldsA[16 * LDS_STRIDE];
  __shared__ float ldsB[16 * LDS_STRIDE];
  const int lane = threadIdx.x;
  const int l15  = lane & 15;
  const int half = lane >> 4;           // 0: K=k0,k0+1   1: K=k0+2,k0+3
  const int r0   = blockIdx.x * 16;

  __builtin_prefetch(W1p + lane * (DIM * DIM / 32), 0, 0);
  __builtin_prefetch(W2p + lane * (DIM * DIM / 32), 0, 0);

#if __has_builtin(__builtin_amdgcn_tensor_load_to_lds)
  {
    // TDM D#: 2D tile [16 rows x 128 f32] from hbuf row r0 into ldsA,
    // pad 2 DWORDs after every 128 DWORDs (LDS row stride = 130 floats).
    unsigned long long ga = (unsigned long long)(uintptr_t)(hbuf + (size_t)r0 * DIM);
    unsigned int lds_off  = (unsigned int)(uintptr_t)(&ldsA[0]);
    v4u g0 = { 1u,                                     // count=1, user mode
               lds_off,                                // lds_addr (bytes)
               (unsigned)(ga & 0xFFFFFFFFu),           // global_addr[31:0]
               (unsigned)((ga >> 32) & 0x01FFFFFFu) | (2u << 30) }; // [56:32]|type=2
    v8i g1 = { (int)((2u << 16)        // data_size = 4B
                     | (1u << 20)      // pad_enable
                     | (6u << 22)      // pad_interval: 128 DWORDs
                     | (1u << 25)),    // pad_amount: 2 DWORDs
               (int)((unsigned)DIM << 16),                       // tensor_dim0 lo16
               (int)(((unsigned)NODES & 0xFFFFu) << 16),         // dim0 hi16 | dim1 lo16
               (int)((((unsigned)NODES >> 16) & 0xFFFFu) | ((unsigned)DIM << 16)), // dim1 hi16 | tile_dim0
               16,                     // tile_dim1 = 16 rows (tile_dim2 = 0)
               DIM,                    // tensor_dim0_stride lo32 = 128
               0, 0 };
    v4i gz4 = { 0, 0, 0, 0 };
    v8i gz8 = { 0, 0, 0, 0, 0, 0, 0, 0 };
    __builtin_amdgcn_tensor_load_to_lds(g0, g1, gz4, gz4, gz8, 0);
    __builtin_amdgcn_s_wait_tensorcnt(0);
  }
#else
  for (int c = lane; c < 16 * DIM; c += 32) {
    int row = c >> 7, q = c & (DIM - 1);
    ldsA[row * LDS_STRIDE + q] = hbuf[(size_t)(r0 + row) * DIM + q];
  }
#endif
  __syncthreads();

  v8f acc[8] = {};
  for (int k0i = 0; k0i < 32; ++k0i) {
    const int ka = k0i * 4 + half * 2;
    v2f a = *(const v2f*)&ldsA[l15 * LDS_STRIDE + ka];
    const float4* Bp = (const float4*)(W1p + ((k0i * 32 + lane) << 4));
    float4 q0 = Bp[0], q1 = Bp[1], q2 = Bp[2], q3 = Bp[3];
    acc[0] = wmma_f32(a, (v2f){q0.x, q0.y}, acc[0]);
    acc[1] = wmma_f32(a, (v2f){q0.z, q0.w}, acc[1]);
    acc[2] = wmma_f32(a, (v2f){q1.x, q1.y}, acc[2]);
    acc[3] = wmma_f32(a, (v2f){q1.z, q1.w}, acc[3]);
    acc[4] = wmma_f32(a, (v2f){q2.x, q2.y}, acc[4]);
    acc[5] = wmma_f32(a, (v2f){q2.z, q2.w}, acc[5]);
    acc[6] = wmma_f32(a, (v2f){q3.x, q3.y}, acc[6]);
    acc[7] = wmma_f32(a, (v2f){q3.z, q3.w}, acc[7]);
  }

  // affine + ReLU, stash intermediate tile in LDS (A-frag layout for GEMM2)
#pragma unroll
  for (int j = 0; j < 8; ++j) {
    const int col = j * 16 + l15;
    const float al = a1[col], sh = s1[col];
#pragma unroll
    for (int p = 0; p < 8; ++p) {
      float v = acc[j][p] * al + sh;              // C frag: M=p+half*8, N=col
      ldsB[(p + half * 8) * LDS_STRIDE + col] = v > 0.f ? v : 0.f;
    }
  }
  __syncthreads();

  v8f acc2[8] = {};
  for (int k0i = 0; k0i < 32; ++k0i) {
    const int ka = k0i * 4 + half * 2;
    v2f a = *(const v2f*)&ldsB[l15 * LDS_STRIDE + ka];
    const float4* Bp = (const float4*)(W2p + ((k0i * 32 + lane) << 4));
    float4 q0 = Bp[0], q1 = Bp[1], q2 = Bp[2], q3 = Bp[3];
    acc2[0] = wmma_f32(a, (v2f){q0.x, q0.y}, acc2[0]);
    acc2[1] = wmma_f32(a, (v2f){q0.z, q0.w}, acc2[1]);
    acc2[2] = wmma_f32(a, (v2f){q1.x, q1.y}, acc2[2]);
    acc2[3] = wmma_f32(a, (v2f){q1.z, q1.w}, acc2[3]);
    acc2[4] = wmma_f32(a, (v2f){q2.x, q2.y}, acc2[4]);
    acc2[5] = wmma_f32(a, (v2f){q2.z, q2.w}, acc2[5]);
    acc2[6] = wmma_f32(a, (v2f){q3.x, q3.y}, acc2[6]);
    acc2[7] = wmma_f32(a, (v2f){q3.z, q3.w}, acc2[7]);
  }

#pragma unroll
  for (int j = 0; j < 8; ++j) {
    const int col = j * 16 + l15;
    const float al = a2[col], sh = s2[col];
#pragma unroll
    for (int p = 0; p < 8; ++p) {
      float v = acc2[j][p] * al + sh;
      v = v > 0.f ? v : 0.f;
      hbuf[(size_t)(r0 + p + half * 8) * DIM + col] = v;
    }
  }
}

__global__ __launch_bounds__(256) void k_pool_init(unsigned* __restrict__ pooled) {
  int i = blockIdx.x * blockDim.x + threadIdx.x;
  if (i < NGRAPH * DIM) pooled[i] = 0u;       // +0.0f bits; h >= 0 post-ReLU
}

// segment_max via u32 atomicMax (valid: post-ReLU values are non-negative)
__global__ __launch_bounds__(256) void k_pool(const float* __restrict__ h,
                                              const int* __restrict__ batch,
                                              unsigned* __restrict__ pooled) {
  int t = blockIdx.x * blockDim.x + threadIdx.x;
  if (t >= NODES * DIM) return;
  int row = t >> 7, col = t & (DIM - 1);
  int g = batch[row];
  atomicMax(&pooled[g * DIM + col], __float_as_uint(h[t]));
}

__global__ __launch_bounds__(64) void k_classify(const unsigned* __restrict__ pooled,
                                                 const float* __restrict__ lw,
                                                 const float* __restrict__ lb,
                                                 float* __restrict__ out) {
  int g = threadIdx.x;
  if (g >= NGRAPH) return;
  float logit[NCLS];
#pragma unroll
  for (int c = 0; c < NCLS; ++c) logit[c] = lb[c];
  for (int d = 0; d < DIM; ++d) {
    float v = __uint_as_float(pooled[g * DIM + d]);
#pragma unroll
    for (int c = 0; c < NCLS; ++c) logit[c] += v * lw[d * NCLS + c];
  }
  float m = logit[0];
#pragma unroll
  for (int c = 1; c < NCLS; ++c) m = fmaxf(m, logit[c]);
  float sum = 0.f;
#pragma unroll
  for (int c = 0; c < NCLS; ++c) sum += expf(logit[c] - m);
  float lse = logf(sum);
#pragma unroll
  for (int c = 0; c < NCLS; ++c) out[g * NCLS + c] = logit[c] - m - lse;
}

extern "C" void kernel_launch(void* const* d_in, const int* in_sizes, int n_in,
                              void* d_out, int out_size, void* d_ws, size_t ws_size,
                              hipStream_t stream) {
  (void)in_sizes; (void)n_in; (void)out_size; (void)ws_size;
  const float* x     = (const float*)d_in[0];
  const int*   ei    = (const int*)d_in[1];     // [2, E]
  const int*   batch = (const int*)d_in[2];     // [N]
  const float* Ws    = (const float*)d_in[3];   // [L,K,D,D]
  const float* bs    = (const float*)d_in[4];
  const float* gam   = (const float*)d_in[5];
  const float* bet   = (const float*)d_in[6];
  const float* rm    = (const float*)d_in[7];
  const float* rv    = (const float*)d_in[8];
  const float* lw    = (const float*)d_in[9];   // [D,C]
  const float* lb    = (const float*)d_in[10];  // [C]
  float* out = (float*)d_out;

  float* bufA  = (float*)d_ws;                          // N*D
  float* bufB  = bufA + (size_t)NODES * DIM;            // N*D
  float* alpha = bufB + (size_t)NODES * DIM;            // L*K*D
  float* shift = alpha + NLAYER * 2 * DIM;              // L*K*D
  float* wpack = shift + NLAYER * 2 * DIM;              // 6 * 16384
  unsigned* pooled = (unsigned*)(wpack + NLAYER * 2 * DIM * DIM); // G*D

  const int* src = ei;
  const int* dst = ei + EDGES;

  k_affine<<<(NLAYER * 2 * DIM + 255) / 256, 256, 0, stream>>>(bs, gam, bet, rm, rv,
                                                               alpha, shift);
  k_packW<<<(NLAYER * 2 * DIM * DIM + 255) / 256, 256, 0, stream>>>(Ws, wpack);
  k_pool_init<<<(NGRAPH * DIM + 255) / 256, 256, 0, stream>>>(pooled);

  const float* hin = x;
  float* cur = bufA;
  float* other = bufB;
  for (int l = 0; l < NLAYER; ++l) {
    k_copy4<<<(NODES * DIM / 4 + 255) / 256, 256, 0, stream>>>(
        (const float4*)hin, (float4*)cur, NODES * DIM / 4);
    k_scatter<<<EDGES / 8, 256, 0, stream>>>(hin, cur, src, dst);
    k_mlp<<<NODES / 16, 32, 0, stream>>>(cur,
                                         wpack + (size_t)(l * 2 + 0) * DIM * DIM,
                                         wpack + (size_t)(l * 2 + 1) * DIM * DIM,
                                         alpha + (l * 2 + 0) * DIM,
                                         shift + (l * 2 + 0) * DIM,
                                         alpha + (l * 2 + 1) * DIM,
                                         shift + (l * 2 + 1) * DIM);
    hin = cur;
    float* t = cur; cur = other; other = t;
  }

  k_pool<<<(NODES * DIM + 255) / 256, 256, 0, stream>>>(hin, batch, pooled);
  k_classify<<<1, 64, 0, stream>>>(pooled, lw, lb, out);
}